// RPE_87565793231069
// MI455X (gfx1250) — compile-verified
//
#include <hip/hip_runtime.h>
#include <math.h>

#define NEG_INF -1000000000.0f

typedef __attribute__((ext_vector_type(4))) float v4f;
typedef __attribute__((ext_vector_type(4))) int   v4i;

// Pointee types in explicit address spaces for the async-to-LDS builtin:
// param0 is int4 in AS1 (global), param1 is int4 in AS3 (LDS).
typedef __attribute__((address_space(1))) v4i as1_v4i;
typedef __attribute__((address_space(3))) v4i as3_v4i;

#define THREADS        256
#define ROWS_PER_BLOCK 8
#define COLS_PER_BLOCK (THREADS * 4)   // 1024 columns, one float4 per lane

// out[i*seq + j] = table[i-j-1] if 1 <= i-j <= maxp-1, else -1e9
__global__ __launch_bounds__(THREADS)
void rpe_fill_kernel(const float* __restrict__ table,
                     float* __restrict__ out,
                     int seq, int maxp)
{
    extern __shared__ float sh[];   // maxp floats (8 KB for maxp=2048)

    const int tid     = threadIdx.x;
    const int row0    = blockIdx.y * ROWS_PER_BLOCK;
    const int colBase = blockIdx.x * COLS_PER_BLOCK;
    const int j0      = colBase + tid * 4;

    // Band for row i: valid columns j in [i-maxp+1, i-1].
    // Union over this block's rows -> cols in [row0-maxp+1, row0+ROWS-2].
    const bool blockTouchesBand =
        (colBase <= row0 + ROWS_PER_BLOCK - 2) &&
        (colBase + COLS_PER_BLOCK - 1 >= row0 - maxp + 1);

    if (blockTouchesBand) {
        // Stage the table into LDS via the gfx1250 async-to-LDS path
        // (ASYNCcnt-tracked copy); fall back to a plain copy if absent.
#if defined(__has_builtin) && __has_builtin(__builtin_amdgcn_global_load_async_to_lds_b128)
        for (int base = tid * 4; base < maxp; base += THREADS * 4) {
            __builtin_amdgcn_global_load_async_to_lds_b128(
                (as1_v4i*)(table + base),
                (as3_v4i*)(sh + base),
                0, 0);
        }
#if __has_builtin(__builtin_amdgcn_s_wait_asynccnt)
        __builtin_amdgcn_s_wait_asynccnt(0);
#else
        asm volatile("s_wait_asynccnt 0" ::: "memory");
#endif
#else
        for (int t = tid; t < maxp; t += THREADS) sh[t] = table[t];
#endif
        __syncthreads();
    }

    if (j0 >= seq) return;

    const v4f neg4 = { NEG_INF, NEG_INF, NEG_INF, NEG_INF };
    const int dmax = maxp - 2;   // valid d = i-j-1 lies in [0, maxp-2]

    #pragma unroll
    for (int r = 0; r < ROWS_PER_BLOCK; ++r) {
        const int i = row0 + r;
        if (i >= seq) break;

        v4f v = neg4;
        // All four lanes outside the band for this row?
        const bool allOut = (!blockTouchesBand) ||
                            (j0 >= i) ||                       // at/above diagonal
                            (j0 + 3 < i - maxp + 1);           // below band
        if (!allOut) {
            #pragma unroll
            for (int k = 0; k < 4; ++k) {
                int d  = i - (j0 + k) - 1;
                int dc = d < 0 ? 0 : (d > dmax ? dmax : d);    // clamp: safe LDS read
                float tv = sh[dc];
                v[k] = ((unsigned)d <= (unsigned)dmax) ? tv : NEG_INF;
            }
        }
        // Output (256 MB) > L2 (192 MB), written once: stream with NT stores.
        __builtin_nontemporal_store(
            v, (v4f*)(out + (size_t)i * (size_t)seq + (size_t)j0));
    }
}

extern "C" void kernel_launch(void* const* d_in, const int* in_sizes, int n_in,
                              void* d_out, int out_size, void* d_ws, size_t ws_size,
                              hipStream_t stream) {
    (void)n_in; (void)d_ws; (void)ws_size;

    const float* table = (const float*)d_in[0];
    const int maxp = in_sizes[0];                        // table rows (2048)
    // Output is [seq, seq, 1] -> seq = sqrt(out_size).
    const int seq = (int)(sqrt((double)out_size) + 0.5);
    float* out = (float*)d_out;

    dim3 block(THREADS);
    dim3 grid((seq + COLS_PER_BLOCK - 1) / COLS_PER_BLOCK,
              (seq + ROWS_PER_BLOCK - 1) / ROWS_PER_BLOCK);
    size_t smem = (size_t)maxp * sizeof(float);

    hipLaunchKernelGGL(rpe_fill_kernel, grid, block, smem, stream,
                       table, out, seq, maxp);
}